// LDIF_87351044866359
// MI455X (gfx1250) — compile-verified
//
#include <hip/hip_runtime.h>
#include <math.h>

#define F_DIM   128
#define EC      32
#define SYM     16
#define E_TOT   48          // EC + SYM
#define B_SZ    4
#define S_SZ    2048
#define NROWS   (B_SZ * E_TOT)   // 192
#define ROWREC  800              // floats per row record in workspace
#define LSTR    132              // padded LDS row stride (floats) -> bank spread
#define MATF    (128 * LSTR)     // floats per padded 128x128 matrix in LDS
#define INV_CBN 0.9999950000375f // 1/sqrt(1 + 1e-5)

typedef float v2f __attribute__((ext_vector_type(2)));
typedef float v8f __attribute__((ext_vector_type(8)));
typedef unsigned int u32x4 __attribute__((ext_vector_type(4)));
typedef int i32x4 __attribute__((ext_vector_type(4)));
typedef int i32x8 __attribute__((ext_vector_type(8)));

// ---------------------------------------------------------------------------
// Kernel 1: per element-row precompute.
//   - 6 CBN vectors: beta/gamma for 3 CBN layers (gamma pre-scaled by 1/sqrt(1+eps))
//   - scalar record: const, center, inv-std, rotation matrix, z-flip
// Row r = b*48 + e ; e >= EC maps to source element (e-EC) with z-mirrored samples.
// ---------------------------------------------------------------------------
__global__ __launch_bounds__(128)
void ldif_prep(const float* __restrict__ pv,   // (B, EC, 138)
               const float* __restrict__ b1w, const float* __restrict__ b1b,
               const float* __restrict__ g1w, const float* __restrict__ g1b,
               const float* __restrict__ b2w, const float* __restrict__ b2b,
               const float* __restrict__ g2w, const float* __restrict__ g2b,
               const float* __restrict__ b3w, const float* __restrict__ b3b,
               const float* __restrict__ g3w, const float* __restrict__ g3b,
               float* __restrict__ ws) {
  __shared__ float emb[F_DIM];
  const int r  = blockIdx.x;
  const int b  = r / E_TOT;
  const int e  = r % E_TOT;
  const int se = (e < EC) ? e : (e - EC);
  const float* row = pv + ((size_t)b * EC + se) * 138;
  const int t = threadIdx.x;

  emb[t] = row[10 + t];
  __syncthreads();

  float* rec = ws + (size_t)r * ROWREC;

  float a1 = b1b[t], a2 = g1b[t], a3 = b2b[t], a4 = g2b[t], a5 = b3b[t], a6 = g3b[t];
  for (int k = 0; k < F_DIM; ++k) {
    const float ek = emb[k];
    a1 = fmaf(ek, b1w[k * F_DIM + t], a1);
    a2 = fmaf(ek, g1w[k * F_DIM + t], a2);
    a3 = fmaf(ek, b2w[k * F_DIM + t], a3);
    a4 = fmaf(ek, g2w[k * F_DIM + t], a4);
    a5 = fmaf(ek, b3w[k * F_DIM + t], a5);
    a6 = fmaf(ek, g3w[k * F_DIM + t], a6);
  }
  rec[0 * F_DIM + t] = a2 * INV_CBN;  // gamma1'
  rec[1 * F_DIM + t] = a1;            // beta1
  rec[2 * F_DIM + t] = a4 * INV_CBN;  // gamma2'
  rec[3 * F_DIM + t] = a3;            // beta2
  rec[4 * F_DIM + t] = a6 * INV_CBN;  // gamma3'
  rec[5 * F_DIM + t] = a5;            // beta3

  if (t == 0) {
    float* sc = rec + 768;
    sc[0] = row[0];                    // const (quad has only [3][3] -> per_element_constants == const)
    sc[1] = row[1]; sc[2] = row[2]; sc[3] = row[3];   // center
    sc[4] = 1.0f / fmaxf(row[4], 1e-4f);
    sc[5] = 1.0f / fmaxf(row[5], 1e-4f);
    sc[6] = 1.0f / fmaxf(row[6], 1e-4f);
    const float rr = row[7], pp = row[8], yy = row[9];
    const float cr = cosf(rr), sr = sinf(rr);
    const float cp = cosf(pp), sp = sinf(pp);
    const float cy = cosf(yy), sy = sinf(yy);
    sc[7 + 0] = cy * cp; sc[7 + 1] = cy * sp * sr - sy * cr; sc[7 + 2] = cy * sp * cr + sy * sr;
    sc[7 + 3] = sy * cp; sc[7 + 4] = sy * sp * sr + cy * cr; sc[7 + 5] = sy * sp * cr - cy * sr;
    sc[7 + 6] = -sp;     sc[7 + 7] = cp * sr;                sc[7 + 8] = cp * cr;
    sc[16] = (e >= EC) ? -1.0f : 1.0f;  // z flip
  }
}

// ---------------------------------------------------------------------------
// TDM: DMA one 128x128 f32 weight matrix from global into LDS, padded to a
// 132-float row stride via D# pad fields (pad_interval=6 -> every 128 DWORDs,
// pad_amount=3 -> 4 DWORDs). 2-D descriptor, groups 2/3 zero (dims 2..4 unused).
// 6-arg builtin (clang-23 therock signature).
// ---------------------------------------------------------------------------
__device__ inline void tdm_load_matrix(const float* __restrict__ gsrc, float* lds_dst) {
  const unsigned long long ga = (unsigned long long)(size_t)gsrc;
  const unsigned lds = (unsigned)(size_t)lds_dst;   // flat LDS addr: low 32 bits = LDS byte offset
  u32x4 g0;
  g0[0] = 1u;                                        // count=1, is_restore=0, gather off
  g0[1] = lds;                                       // lds_addr (bytes)
  g0[2] = (unsigned)(ga & 0xFFFFFFFFu);              // global_addr[31:0]
  g0[3] = (unsigned)((ga >> 32) & 0x01FFFFFFu) |     // global_addr[56:32]
          (2u << 30);                                // type=2 ("image")
  i32x8 g1;
  g1[0] = (int)((2u << 16) |                         // data_size = 4B
                (1u << 20) |                         // pad_enable
                (6u << 22) |                         // pad_interval: 128 DWORDs
                (3u << 25));                         // pad_amount:  4 DWORDs
  g1[1] = (int)(128u << 16);                         // tensor_dim0[15:0] at bits 63:48
  g1[2] = (int)(128u << 16);                         // tensor_dim1[15:0] at bits 95:80
  g1[3] = (int)(128u << 16);                         // tile_dim0 = 128 (bits 127:112)
  g1[4] = (int)128;                                  // tile_dim1 = 128, tile_dim2 = 0
  g1[5] = (int)128;                                  // tensor_dim0_stride = 128 elements
  g1[6] = 0;                                         // stride hi / tensor_dim1_stride lo
  g1[7] = 0;
  const i32x4 gz4 = {0, 0, 0, 0};                    // groups 2/3: dims 2..4 unused
  const i32x8 gz8 = {0, 0, 0, 0, 0, 0, 0, 0};
  __builtin_amdgcn_tensor_load_to_lds(g0, g1, gz4, gz4, gz8, 0);
}

// ---------------------------------------------------------------------------
// 128x128x128 GEMM building block, fp32 WMMA 16x16x4, A and B both in LDS.
// One wave: M-strip of 16 rows (m0), all 8 N-tiles accumulated in acc[8].
// A layout: lanes 0-15 -> {K=k0,k0+1}, lanes 16-31 -> {K=k0+2,k0+3} (ISA 16x4 f32).
// B layout mirrors it across the two lane halves.
// ---------------------------------------------------------------------------
__device__ inline void gemm128(const float* Abuf, bool relu_a,
                               const float* Wlds,
                               int m0, int lrow, int khalf, v8f acc[8]) {
  for (int kk = 0; kk < 32; ++kk) {
    const int ka = kk * 4 + khalf;
    float a0 = Abuf[(m0 + lrow) * LSTR + ka];
    float a1 = Abuf[(m0 + lrow) * LSTR + ka + 1];
    if (relu_a) { a0 = fmaxf(a0, 0.0f); a1 = fmaxf(a1, 0.0f); }
    v2f a; a.x = a0; a.y = a1;
#pragma unroll
    for (int nt = 0; nt < 8; ++nt) {
      const int col = nt * 16 + lrow;
      v2f bf;
      bf.x = Wlds[ka * LSTR + col];
      bf.y = Wlds[(ka + 1) * LSTR + col];
      acc[nt] = __builtin_amdgcn_wmma_f32_16x16x4_f32(
          false, a, false, bf, (short)0, acc[nt], false, false);
    }
  }
}

// ---------------------------------------------------------------------------
// Kernel 2: one block = one element-row x 128 samples. 256 threads (8 waves).
// LDS: wbuf1 | wbuf2 | ibuf | hbuf (each 128x132 f32) + xbuf (128x4).
// ---------------------------------------------------------------------------
__global__ __launch_bounds__(256)
void ldif_main(const float* __restrict__ samples,  // (B, S, 3)
               const float* __restrict__ fc1_w, const float* __restrict__ fc1_b,
               const float* __restrict__ rfc1_w, const float* __restrict__ rfc1_b,
               const float* __restrict__ rfc2_w, const float* __restrict__ rfc2_b,
               const float* __restrict__ fc2_w, const float* __restrict__ fc2_b,
               const float* __restrict__ ws_rows,
               float* __restrict__ partial) {
  extern __shared__ float smem[];
  float* wbuf1 = smem;                 // rfc1_w, padded stride 132
  float* wbuf2 = smem + 1 * MATF;      // rfc2_w
  float* ibuf  = smem + 2 * MATF;      // init_h (CBN1 output)
  float* hbuf  = smem + 3 * MATF;      // intermediate / final h
  float* xbuf  = smem + 4 * MATF;      // 128 x {x0,x1,x2,wexp}

  const int r    = blockIdx.x;
  const int tile = blockIdx.y;
  const int b    = r / E_TOT;
  const float* rec = ws_rows + (size_t)r * ROWREC;
  const float* sc  = rec + 768;
  const int t = threadIdx.x;

  // ---- Kick off weight DMA (wave 0 only; TDM ignores EXEC, TENSORcnt per-wave) ----
  if (t < 32) {
    tdm_load_matrix(rfc1_w, wbuf1);
    tdm_load_matrix(rfc2_w, wbuf2);
  }

  // ---- Phase A: geometry -> local_samples + gaussian weight (overlaps DMA) ----
  if (t < 128) {
    const int s = tile * 128 + t;
    const float* p = samples + ((size_t)b * S_SZ + s) * 3;
    const float d0 = p[0]          - sc[1];
    const float d1 = p[1]          - sc[2];
    const float d2 = p[2] * sc[16] - sc[3];
    const float l0 = d0 * sc[7 + 0] + d1 * sc[7 + 3] + d2 * sc[7 + 6];
    const float l1 = d0 * sc[7 + 1] + d1 * sc[7 + 4] + d2 * sc[7 + 7];
    const float l2 = d0 * sc[7 + 2] + d1 * sc[7 + 5] + d2 * sc[7 + 8];
    const float x0 = l0 * sc[4], x1 = l1 * sc[5], x2 = l2 * sc[6];
    xbuf[t * 4 + 0] = x0;
    xbuf[t * 4 + 1] = x1;
    xbuf[t * 4 + 2] = x2;
    xbuf[t * 4 + 3] = expf(-0.5f * (x0 * x0 + x1 * x1 + x2 * x2));
  }
  __syncthreads();

  // ---- Phase B: init_h = CBN1(fc1(x)) -> ibuf (overlaps DMA) ----
  {
    const int n = t & 127;
    const int sbase = (t >> 7) * 64;
    const float w0 = fc1_w[0 * F_DIM + n];
    const float w1 = fc1_w[1 * F_DIM + n];
    const float w2 = fc1_w[2 * F_DIM + n];
    const float bias = fc1_b[n];
    const float g1 = rec[0 * F_DIM + n], be1 = rec[1 * F_DIM + n];
    for (int s = sbase; s < sbase + 64; ++s) {
      float h = fmaf(xbuf[s * 4 + 0], w0,
               fmaf(xbuf[s * 4 + 1], w1,
               fmaf(xbuf[s * 4 + 2], w2, bias)));
      ibuf[s * LSTR + n] = fmaf(g1, h, be1);
    }
  }

  // ---- Publish DMA'd weights: issuing wave waits, barrier releases the rest ----
  if (t < 32) {
    __builtin_amdgcn_s_wait_tensorcnt(0);
  }
  __syncthreads();

  const int wave  = t >> 5;
  const int lane  = t & 31;
  const int m0    = wave * 16;
  const int lrow  = lane & 15;
  const int khalf = (lane >> 4) * 2;
  const int mhalf = (lane >> 4) ? 8 : 0;

  // ---- GEMM1: H2 = relu(init_h) @ rfc1_w ----
  v8f acc[8] = {};
  gemm128(ibuf, /*relu_a=*/true, wbuf1, m0, lrow, khalf, acc);

  // epilogue 1: hbuf = relu(CBN2(H2 + rfc1_b))
#pragma unroll
  for (int nt = 0; nt < 8; ++nt) {
    const int col = nt * 16 + lrow;
    const float g = rec[2 * F_DIM + col], be = rec[3 * F_DIM + col], bs = rfc1_b[col];
#pragma unroll
    for (int j = 0; j < 8; ++j) {
      const int m = m0 + j + mhalf;
      float v = acc[nt][j] + bs;
      v = fmaf(g, v, be);
      hbuf[m * LSTR + col] = fmaxf(v, 0.0f);
    }
  }
  __syncthreads();

  // ---- GEMM2: H3 = hbuf @ rfc2_w ----
  v8f acc2[8] = {};
  gemm128(hbuf, /*relu_a=*/false, wbuf2, m0, lrow, khalf, acc2);
  __syncthreads();   // everyone done reading hbuf before we overwrite it

  // epilogue 2: hbuf = CBN3(init_h + H3 + rfc2_b)
#pragma unroll
  for (int nt = 0; nt < 8; ++nt) {
    const int col = nt * 16 + lrow;
    const float g = rec[4 * F_DIM + col], be = rec[5 * F_DIM + col], bs = rfc2_b[col];
#pragma unroll
    for (int j = 0; j < 8; ++j) {
      const int m = m0 + j + mhalf;
      float v = acc2[nt][j] + bs + ibuf[m * LSTR + col];
      hbuf[m * LSTR + col] = fmaf(g, v, be);
    }
  }
  __syncthreads();

  // ---- Phase E: fc2 + residual + local decision ----
  if (t < 128) {
    float val = fc2_b[0];
    for (int n = 0; n < F_DIM; ++n)
      val = fmaf(hbuf[t * LSTR + n], fc2_w[n], val);
    const float dec = sc[0] * xbuf[t * 4 + 3] * (1.0f + val);
    partial[(size_t)r * S_SZ + tile * 128 + t] = dec;
  }
}

// ---------------------------------------------------------------------------
// Kernel 3: sdf[b][s] = sum_e partial[(b*48+e)][s]   (deterministic reduce)
// ---------------------------------------------------------------------------
__global__ void ldif_reduce(const float* __restrict__ partial, float* __restrict__ out) {
  const int i = blockIdx.x * blockDim.x + threadIdx.x;
  if (i >= B_SZ * S_SZ) return;
  const int b = i / S_SZ, s = i % S_SZ;
  float acc = 0.0f;
  for (int e = 0; e < E_TOT; ++e)
    acc += partial[((size_t)(b * E_TOT + e)) * S_SZ + s];
  out[i] = acc;
}

// ---------------------------------------------------------------------------
extern "C" void kernel_launch(void* const* d_in, const int* in_sizes, int n_in,
                              void* d_out, int out_size, void* d_ws, size_t ws_size,
                              hipStream_t stream) {
  const float* pv      = (const float*)d_in[0];
  const float* samples = (const float*)d_in[1];
  const float* fc1_w   = (const float*)d_in[2];
  const float* fc1_b   = (const float*)d_in[3];
  const float* b1w     = (const float*)d_in[4];
  const float* b1b     = (const float*)d_in[5];
  const float* g1w     = (const float*)d_in[6];
  const float* g1b     = (const float*)d_in[7];
  const float* rfc1_w  = (const float*)d_in[8];
  const float* rfc1_b  = (const float*)d_in[9];
  const float* b2w     = (const float*)d_in[10];
  const float* b2b     = (const float*)d_in[11];
  const float* g2w     = (const float*)d_in[12];
  const float* g2b     = (const float*)d_in[13];
  const float* rfc2_w  = (const float*)d_in[14];
  const float* rfc2_b  = (const float*)d_in[15];
  const float* b3w     = (const float*)d_in[16];
  const float* b3b     = (const float*)d_in[17];
  const float* g3w     = (const float*)d_in[18];
  const float* g3b     = (const float*)d_in[19];
  const float* fc2_w   = (const float*)d_in[20];
  const float* fc2_b   = (const float*)d_in[21];

  float* rowrec  = (float*)d_ws;                       // 192*800 floats
  float* partial = rowrec + (size_t)NROWS * ROWREC;    // 192*2048 floats

  const size_t lds_bytes = (size_t)(4 * MATF + 128 * 4) * sizeof(float); // ~266 KB
  (void)hipFuncSetAttribute((const void*)ldif_main,
                            hipFuncAttributeMaxDynamicSharedMemorySize,
                            (int)lds_bytes);

  ldif_prep<<<dim3(NROWS), dim3(128), 0, stream>>>(
      pv, b1w, b1b, g1w, g1b, b2w, b2b, g2w, g2b, b3w, b3b, g3w, g3b, rowrec);

  ldif_main<<<dim3(NROWS, S_SZ / 128), dim3(256), lds_bytes, stream>>>(
      samples, fc1_w, fc1_b, rfc1_w, rfc1_b, rfc2_w, rfc2_b, fc2_w, fc2_b,
      rowrec, partial);

  ldif_reduce<<<dim3((B_SZ * S_SZ + 255) / 256), dim3(256), 0, stream>>>(
      partial, (float*)d_out);
}